// GATAttentionHead_38835094290620
// MI455X (gfx1250) — compile-verified
//
#include <hip/hip_runtime.h>
#include <hip/hip_bf16.h>

typedef __attribute__((ext_vector_type(2))) float v2f;
typedef __attribute__((ext_vector_type(8))) float v8f;

#define NEG_SLOPE 0.2f
#define IN_DIM 256
#define OUT_DIM 64

// ---------------------------------------------------------------------------
// Kernel 1: h = x @ W^T via V_WMMA_F32_16X16X4_F32 (wave32), fused s1/s2.
// One wave computes a 16x64 tile of h.
// LDS holds W^T in k-pair-interleaved form: lds[(k>>1)*128 + n*2 + (k&1)],
// so a lane's B fragment (k, k+1) is one contiguous ds_load_b64 straight into
// the even-aligned VGPR pair the WMMA consumes (no repack movs).
// ---------------------------------------------------------------------------
__global__ __launch_bounds__(128) void gat_gemm_kernel(
    const float* __restrict__ x, const float* __restrict__ W,
    const float* __restrict__ a1, const float* __restrict__ a2,
    float* __restrict__ h, float* __restrict__ s1, float* __restrict__ s2,
    int N)
{
    __shared__ float ldsWT[IN_DIM * OUT_DIM];  // 64 KB, [k/2][n][2]

    const int tid = threadIdx.x;
    // Cooperative fill: coalesced float2 reads of W [n][k] -> interleaved LDS
    for (int idx2 = tid; idx2 < (IN_DIM / 2) * OUT_DIM; idx2 += 128) {
        const int n = idx2 >> 7;       // / (IN_DIM/2)
        const int p = idx2 & 127;      // k pair index
        const v2f w2 = ((const v2f*)W)[(size_t)n * (IN_DIM / 2) + p];
        *(v2f*)&ldsWT[(p * OUT_DIM + n) * 2] = w2;
    }
    __syncthreads();

    const int wave  = blockIdx.x * 4 + (tid >> 5);
    const int lane  = tid & 31;
    const int mlane = lane & 15;
    const int khalf = (lane >> 4) << 1;          // 0 or 2 (A/B K split by lane half)
    const int tiles = (N + 15) >> 4;
    if (wave >= tiles) return;                   // wave-uniform: EXEC stays all-1s

    const int row_base = wave << 4;
    int row = row_base + mlane;
    if (row > N - 1) row = N - 1;                // clamp loads (stores are guarded)
    const float* xrow = x + (size_t)row * IN_DIM;

    v8f acc[4] = {};
    for (int k = 0; k < IN_DIM; k += 4) {
        const int kk = k + khalf;                // even
        const v2f a = *(const v2f*)&xrow[kk];    // one global_load_b64
        const float* brow = &ldsWT[(kk >> 1) * (OUT_DIM * 2)];
        #pragma unroll
        for (int t = 0; t < 4; ++t) {
            const v2f b = *(const v2f*)&brow[(t * 16 + mlane) * 2]; // ds_load_b64
            acc[t] = __builtin_amdgcn_wmma_f32_16x16x4_f32(
                false, a, false, b, (short)0, acc[t], false, false);
        }
    }

    // Epilogue: store h, and reduce s1 = h.a1, s2 = h.a2 across the N dim.
    float a1v[4], a2v[4];
    #pragma unroll
    for (int t = 0; t < 4; ++t) {
        a1v[t] = a1[t * 16 + mlane];
        a2v[t] = a2[t * 16 + mlane];
    }
    const int rowhi = (lane >> 4) << 3;          // 0 or 8
    #pragma unroll
    for (int r = 0; r < 8; ++r) {
        const int  orow  = row_base + rowhi + r;
        const bool valid = orow < N;
        float p1 = 0.f, p2 = 0.f;
        #pragma unroll
        for (int t = 0; t < 4; ++t) {
            const float v = acc[t][r];
            if (valid) h[(size_t)orow * OUT_DIM + t * 16 + mlane] = v;
            p1 += v * a1v[t];
            p2 += v * a2v[t];
        }
        // all-reduce within each 16-lane half (xor masks stay inside the half)
        #pragma unroll
        for (int off = 8; off > 0; off >>= 1) {
            p1 += __shfl_xor(p1, off, 32);
            p2 += __shfl_xor(p2, off, 32);
        }
        if (valid && mlane == 0) { s1[orow] = p1; s2[orow] = p2; }
    }
}

// ---------------------------------------------------------------------------
// Kernel 0: zero out[], init segment max to -inf, segment sum to 0.
// ---------------------------------------------------------------------------
__global__ void gat_init_kernel(float* __restrict__ out, float* __restrict__ mx,
                                float* __restrict__ z, int N)
{
    const int i = blockIdx.x * 256 + threadIdx.x;
    if (i < N * OUT_DIM) out[i] = 0.f;
    if (i < N) { mx[i] = -__builtin_inff(); z[i] = 0.f; }
}

// ---------------------------------------------------------------------------
// Kernel 2: per-edge score = leaky_relu(s1[src]+s2[dst]); segment max over src
// via signed-max / unsigned-min float-bit trick (monotone for IEEE floats).
// Edges [0,E) from edge_index, [E,TE) are self loops.
// ---------------------------------------------------------------------------
__global__ void gat_score_kernel(const int* __restrict__ ei,
                                 const float* __restrict__ s1,
                                 const float* __restrict__ s2,
                                 float* __restrict__ se, float* __restrict__ mx,
                                 int E, int TE)
{
    const int i = blockIdx.x * 256 + threadIdx.x;
    if (i >= TE) return;
    int src, dst;
    if (i < E) { src = ei[i]; dst = ei[E + i]; } else { src = dst = i - E; }
    float v = s1[src] + s2[dst];
    v = (v > 0.f) ? v : NEG_SLOPE * v;
    se[i] = v;
    if (v >= 0.f) atomicMax((int*)(mx + src), __float_as_int(v));
    else          atomicMin((unsigned int*)(mx + src), __float_as_uint(v));
}

// ---------------------------------------------------------------------------
// Kernel 3: e = exp(s - m[src]) (in place over se), z[src] += e.
// ---------------------------------------------------------------------------
__global__ void gat_exp_kernel(const int* __restrict__ ei,
                               const float* __restrict__ mx,
                               float* __restrict__ se, float* __restrict__ z,
                               int E, int TE)
{
    const int i = blockIdx.x * 256 + threadIdx.x;
    if (i >= TE) return;
    const int src = (i < E) ? ei[i] : (i - E);
    const float e = __expf(se[i] - mx[src]);
    se[i] = e;
    atomicAdd(z + src, e);
}

// ---------------------------------------------------------------------------
// Kernel 4: out[src] += (e / z[src]) * h[dst].  One wave per edge; each lane
// loads a float2 of h (global_load_b64) and issues two f32 atomics.  h and
// out (25.6 MB each) are L2-resident on MI455X (192 MB L2).
// ---------------------------------------------------------------------------
__global__ __launch_bounds__(256) void gat_aggr_kernel(
    const int* __restrict__ ei, const float* __restrict__ se,
    const float* __restrict__ z, const float* __restrict__ h,
    float* __restrict__ out, int E, int TE)
{
    const int wave = blockIdx.x * 8 + (threadIdx.x >> 5);
    const int lane = threadIdx.x & 31;
    if (wave >= TE) return;
    int src, dst;
    if (wave < E) { src = ei[wave]; dst = ei[E + wave]; } else { src = dst = wave - E; }
    const float att = se[wave] / z[src];
    const v2f hv = ((const v2f*)(h + (size_t)dst * OUT_DIM))[lane];
    float* orow = out + (size_t)src * OUT_DIM + lane * 2;
    atomicAdd(orow,     att * hv.x);
    atomicAdd(orow + 1, att * hv.y);
}

// ---------------------------------------------------------------------------
extern "C" void kernel_launch(void* const* d_in, const int* in_sizes, int n_in,
                              void* d_out, int out_size, void* d_ws, size_t ws_size,
                              hipStream_t stream)
{
    const float* x  = (const float*)d_in[0];
    const int*   ei = (const int*)d_in[1];   // edge_index [2][E]
    const float* W  = (const float*)d_in[2];
    const float* a1 = (const float*)d_in[3];
    const float* a2 = (const float*)d_in[4];
    const int N  = in_sizes[0] / IN_DIM;
    const int E  = in_sizes[1] / 2;
    const int TE = E + N;
    float* out = (float*)d_out;

    // Workspace layout (floats): h[N*64] | s1[N] | s2[N] | mx[N] | z[N] | se[TE]
    float* h  = (float*)d_ws;
    float* s1 = h  + (size_t)N * OUT_DIM;
    float* s2 = s1 + N;
    float* mx = s2 + N;
    float* z  = mx + N;
    float* se = z  + N;

    gat_init_kernel<<<(N * OUT_DIM + 255) / 256, 256, 0, stream>>>(out, mx, z, N);

    const int tiles = (N + 15) / 16;
    gat_gemm_kernel<<<(tiles + 3) / 4, 128, 0, stream>>>(x, W, a1, a2, h, s1, s2, N);

    gat_score_kernel<<<(TE + 255) / 256, 256, 0, stream>>>(ei, s1, s2, se, mx, E, TE);
    gat_exp_kernel  <<<(TE + 255) / 256, 256, 0, stream>>>(ei, mx, se, z, E, TE);
    gat_aggr_kernel <<<(TE + 7) / 8,     256, 0, stream>>>(ei, se, z, h, out, E, TE);
}